// Quantizer_10548439679060
// MI455X (gfx1250) — compile-verified
//
#include <hip/hip_runtime.h>
#include <hip/hip_bf16.h>
#include <stdint.h>

// VQ-VAE quantizer for MI455X (gfx1250, wave32).
// - Distance GEMM (65536x1024, K=64) via fp32 WMMA (v_wmma_f32_16x16x4_f32),
//   dual accumulators per wave so each LDS B-fragment feeds two WMMAs.
// - Codebook staged into LDS (~278 KB of the 320 KB WGP budget) by the Tensor
//   Data Mover: one D# per workgroup; the 64-dword-data + 4-dword-pad row
//   layout (stride 68) is produced by TDM padding itself.

typedef __attribute__((ext_vector_type(2))) float        v2f;
typedef __attribute__((ext_vector_type(8))) float        v8f;
typedef __attribute__((ext_vector_type(4))) unsigned int v4u;
typedef __attribute__((ext_vector_type(8))) int          v8i;
typedef __attribute__((ext_vector_type(4))) int          v4i;

#define K_CODES    1024
#define DIMS       64
#define EMB_STRIDE 68            // 64 data dwords + 4 pad dwords: 16B-aligned rows
#define QV_SIZE    4194304       // 64*64*32*32
#define IDX_OFF    4194304
#define LOSS_OFF   4259840       // QV_SIZE + 65536
#define LOSS_SCALE (1.25f / 4194304.0f)   // (0.25+1.0) * mean over B*H*W*D

#if defined(__gfx1250__) && __has_builtin(__builtin_amdgcn_tensor_load_to_lds)
#define USE_TDM 1
#else
#define USE_TDM 0
#warning "CDNA5 TDM builtin not available: falling back to vector-load staging"
#endif

__global__ void vq_init_loss(float* out) { out[LOSS_OFF] = 0.0f; }

__global__ __launch_bounds__(256)
void vq_wmma_kernel(const float* __restrict__ z, const float* __restrict__ emb,
                    float* __restrict__ out) {
  extern __shared__ float emb_s[];           // K_CODES * EMB_STRIDE floats
  __shared__ float enorm_s[K_CODES];
  __shared__ int   closest_s[256];           // 8 waves * 32 rows

  const int tid  = threadIdx.x;
  const int wave = tid >> 5, lane = tid & 31;
  const int m = lane & 15, h = lane >> 4;           // WMMA lane coords
  const int r0 = blockIdx.x * 256 + wave * 32;      // two 16-row tiles per wave

  // Warm L2/L0 for this lane's A rows while the codebook staging runs.
  __builtin_prefetch(z + (size_t)(r0 + m) * DIMS, 0, 0);
  __builtin_prefetch(z + (size_t)(r0 + 16 + m) * DIMS, 0, 0);

  // ---- stage codebook into LDS ----
#if USE_TDM
  if (tid < 32) {                                   // wave 0 issues one TDM op
    const uint64_t ga = (uint64_t)(uintptr_t)emb;
    v4u g0;
    g0.x = 1u;                                      // count=1 valid user D#
    g0.y = (unsigned)(uintptr_t)(&emb_s[0]);        // lds_addr (bytes)
    g0.z = (unsigned)(ga & 0xFFFFFFFFu);            // global_addr[31:0]
    g0.w = (unsigned)((ga >> 32) & 0x01FFFFFFu)     // global_addr[56:32]
         | (2u << 30);                              // type=2 ("image")
    v8i g1;
    g1[0] = (2 << 16)                               // data_size: 4 bytes
          | (1 << 20)                               // pad_enable
          | (5 << 22)                               // pad_interval: 64 dwords
          | (3 << 25);                              // pad_amount: 4 dwords
    g1[1] = (int)(64u << 16);                       // tensor_dim0 = 64   (bit48)
    g1[2] = (int)(1024u << 16);                     // tensor_dim1 = 1024 (bit80)
    g1[3] = (int)(64u << 16);                       // tile_dim0 = 64     (bit112)
    g1[4] = 1024;                                   // tile_dim1 = 1024, tile_dim2 = 0
    g1[5] = 64;                                     // tensor_dim0_stride = 64
    g1[6] = (int)(64u << 16);                       // tensor_dim1_stride = 64 (bit208)
    g1[7] = 0;
    v4i g2 = {0, 0, 0, 0};                          // 2-D tensor: groups 2/3 unused
    v4i g3 = {0, 0, 0, 0};
#if (__clang_major__ >= 23)
    v8i g4 = {0, 0, 0, 0, 0, 0, 0, 0};              // clang-23 6-arg form
    __builtin_amdgcn_tensor_load_to_lds(g0, g1, g2, g3, g4, 0);
#else
    __builtin_amdgcn_tensor_load_to_lds(g0, g1, g2, g3, 0);
#endif
#if __has_builtin(__builtin_amdgcn_s_wait_tensorcnt)
    __builtin_amdgcn_s_wait_tensorcnt(0);
#else
    asm volatile("s_wait_tensorcnt 0x0" ::: "memory");
#endif
  }
#else
  for (int c = tid; c < K_CODES * (DIMS / 4); c += 256) {
    const int row = c >> 4, part = c & 15;
    const float4 v = *(const float4*)(emb + row * DIMS + part * 4);
    *(float4*)(&emb_s[row * EMB_STRIDE + part * 4]) = v;
  }
#endif
  __syncthreads();

  // ---- codebook row norms ----
  for (int k = tid; k < K_CODES; k += 256) {
    float s = 0.f;
    #pragma unroll 8
    for (int d = 0; d < DIMS; ++d) { const float e = emb_s[k * EMB_STRIDE + d]; s += e * e; }
    enorm_s[k] = s;
  }
  __syncthreads();

  // ---- A fragments: lane holds row (tile + m), dims 4*kk + 2h + {0,1} ----
  v2f a[2][16];
  #pragma unroll
  for (int t = 0; t < 2; ++t)
    #pragma unroll
    for (int kk = 0; kk < 16; ++kk)
      a[t][kk] = *(const v2f*)(z + (size_t)(r0 + t * 16 + m) * DIMS + kk * 4 + 2 * h);

  float minv[2][8];
  int   mini[2][8];
  #pragma unroll
  for (int t = 0; t < 2; ++t)
    #pragma unroll
    for (int v = 0; v < 8; ++v) { minv[t][v] = 3.4e38f; mini[t][v] = 0; }

  // ---- sweep codebook: 64 N-tiles x (16 chained 16x16x4 fp32 WMMAs) ----
  for (int nt = 0; nt < 64; ++nt) {
    const int code = nt * 16 + m;                    // this lane's column (code)
    v2f bf[16];
    #pragma unroll
    for (int kk = 0; kk < 16; ++kk)                  // B frag: code row, dims 4kk+2h+{0,1}
      bf[kk] = *(const v2f*)(&emb_s[code * EMB_STRIDE + kk * 4 + 2 * h]);

    v8f c0 = {}; v8f c1 = {};
    #pragma unroll
    for (int kk = 0; kk < 16; ++kk) {
      c0 = __builtin_amdgcn_wmma_f32_16x16x4_f32(false, a[0][kk], false, bf[kk],
                                                 (short)0, c0, false, false);
      c1 = __builtin_amdgcn_wmma_f32_16x16x4_f32(false, a[1][kk], false, bf[kk],
                                                 (short)0, c1, false, false);
    }
    const float en = enorm_s[code];
    #pragma unroll
    for (int v = 0; v < 8; ++v) {                    // score = ||e||^2 - 2 z.e
      const float s0 = en - 2.0f * c0[v];
      if (s0 < minv[0][v]) { minv[0][v] = s0; mini[0][v] = code; }
      const float s1 = en - 2.0f * c1[v];
      if (s1 < minv[1][v]) { minv[1][v] = s1; mini[1][v] = code; }
    }
  }

  // ---- cross-lane argmin (butterfly within each 16-lane half), stash in LDS ----
  #pragma unroll
  for (int t = 0; t < 2; ++t) {
    #pragma unroll
    for (int v = 0; v < 8; ++v) {
      float val = minv[t][v]; int idx = mini[t][v];
      #pragma unroll
      for (int mask = 1; mask <= 8; mask <<= 1) {
        const float ov = __shfl_xor(val, mask, 32);
        const int   oi = __shfl_xor(idx, mask, 32);
        if (ov < val || (ov == val && oi < idx)) { val = ov; idx = oi; }
      }
      mini[t][v] = idx;
    }
    if (lane == 0) {
      #pragma unroll
      for (int v = 0; v < 8; ++v) closest_s[wave * 32 + t * 16 + v] = mini[t][v];
    }
    if (lane == 16) {
      #pragma unroll
      for (int v = 0; v < 8; ++v) closest_s[wave * 32 + t * 16 + 8 + v] = mini[t][v];
    }
  }
  __syncthreads();

  // ---- gather winners, transpose-scatter [B,D,H,W], indices, loss ----
  float sse = 0.f;
  #pragma unroll
  for (int t = 0; t < 2; ++t) {
    const int r  = r0 + t * 16 + m;
    const int cm = closest_s[wave * 32 + t * 16 + m];
    const int b  = r >> 10;
    const int hw = r & 1023;
    float* qv = out + (size_t)b * 65536 + hw;
    #pragma unroll
    for (int kk = 0; kk < 16; ++kk) {
      #pragma unroll
      for (int j = 0; j < 2; ++j) {
        const int d = kk * 4 + 2 * h + j;            // reuse A regs as z values
        const float q = emb_s[cm * EMB_STRIDE + d];
        const float diff = q - a[t][kk][j];
        sse += diff * diff;
        qv[d * 1024] = q;                            // lanes 0-15: 16 consecutive floats
      }
    }
    if (h == 0) out[IDX_OFF + r] = (float)cm;
  }
  #pragma unroll
  for (int mask = 1; mask < 32; mask <<= 1) sse += __shfl_xor(sse, mask, 32);
  if (lane == 0) atomicAdd(out + LOSS_OFF, sse * LOSS_SCALE);
}

extern "C" void kernel_launch(void* const* d_in, const int* in_sizes, int n_in,
                              void* d_out, int out_size, void* d_ws, size_t ws_size,
                              hipStream_t stream) {
  const float* z   = (const float*)d_in[0];   // [64,32,32,64] f32
  const float* emb = (const float*)d_in[1];   // [1024,64] f32
  float* out = (float*)d_out;                 // [qv | idx | loss]

  vq_init_loss<<<1, 1, 0, stream>>>(out);
  const int lds_bytes = K_CODES * EMB_STRIDE * (int)sizeof(float);  // ~278 KB (<320 KB WGP)
  vq_wmma_kernel<<<256, 256, lds_bytes, stream>>>(z, emb, out);
}